// Decoder_88734024336146
// MI455X (gfx1250) — compile-verified
//
#include <hip/hip_runtime.h>
#include <hip/hip_bf16.h>
#include <math.h>

// Problem constants (from reference): VOCAB, D, H, B, T, MAXLEN
#define VOCAB 32000
#define DIM   512
#define HID   512
#define BATCH 32
#define TLEN  128
#define LMAX  32
#define DH    1024   // D + H
#define G3    1536   // 3*H

typedef float v2f __attribute__((ext_vector_type(2)));
typedef float v8f __attribute__((ext_vector_type(8)));

// ---------------------------------------------------------------------------
// WMMA helpers: V_WMMA_F32_16X16X4_F32, wave32.
// A-frag (16x4): lane m=l&15, K-pair = 2*(l>>4); float2 load {k, k+1}.
// B-frag (4x16): lane n=l&15, same K-pair mapping.
// C/D (16x16 f32, 8 VGPRs): VGPR i holds M=(l>>4)*8+i, N=l&15.
// ---------------------------------------------------------------------------
__device__ __forceinline__ v8f wmma4(v2f a, v2f b, v8f c) {
    return __builtin_amdgcn_wmma_f32_16x16x4_f32(
        /*neg_a=*/false, a, /*neg_b=*/false, b,
        /*c_mod=*/(short)0, c, /*reuse_a=*/false, /*reuse_b=*/false);
}

__device__ __forceinline__ void store_tile16(float* out, int stride, v8f c, int lane, int row_off) {
    int n  = lane & 15;
    int mg = (lane >> 4) * 8;
#pragma unroll
    for (int i = 0; i < 8; ++i) out[(row_off + mg + i) * (long)stride + n] = c[i];
}

// ---------------------------------------------------------------------------
// One-time: energies[bt, g] = enc[bt,:] . W_attn[g,:] + b_attn[g]
// M=4096 (B*T), N=512, K=512.  Wave tile = 32x32 (4 accumulators):
// 128 M-tiles x 16 N-tiles = 2048 wave-tiles; 8 waves/block -> 256 blocks.
// ---------------------------------------------------------------------------
__global__ void k_energies(const float* __restrict__ enc,
                           const float* __restrict__ Wa,
                           const float* __restrict__ ba,
                           float* __restrict__ energies) {
    int wave = threadIdx.x >> 5, lane = threadIdx.x & 31;
    int tile = blockIdx.x * 8 + wave;          // 0..2047
    int mt = tile >> 4, nt = tile & 15;        // mt: 32-row tile, nt: 32-col tile
    const float* A  = enc + (long)(mt * 32) * HID;
    const float* W0 = Wa  + (long)(nt * 32) * HID;
    const float* W1 = W0 + 16 * HID;
    int r = lane & 15, kh = (lane >> 4) * 2;
    v8f c00 = {}, c01 = {}, c10 = {}, c11 = {};
#pragma unroll 4
    for (int k = 0; k < HID; k += 4) {
        v2f a0 = *(const v2f*)(A  + r * HID + k + kh);
        v2f a1 = *(const v2f*)(A  + (r + 16) * HID + k + kh);
        v2f b0 = *(const v2f*)(W0 + r * HID + k + kh);
        v2f b1 = *(const v2f*)(W1 + r * HID + k + kh);
        c00 = wmma4(a0, b0, c00);
        c10 = wmma4(a1, b0, c10);
        c01 = wmma4(a0, b1, c01);
        c11 = wmma4(a1, b1, c11);
    }
    float bv0 = ba[nt * 32 + r];
    float bv1 = ba[nt * 32 + 16 + r];
#pragma unroll
    for (int i = 0; i < 8; ++i) { c00[i] += bv0; c10[i] += bv0; c01[i] += bv1; c11[i] += bv1; }
    float* O = energies + (long)(mt * 32) * HID + nt * 32;
    store_tile16(O,      HID, c00, lane, 0);
    store_tile16(O,      HID, c10, lane, 16);
    store_tile16(O + 16, HID, c01, lane, 0);
    store_tile16(O + 16, HID, c11, lane, 16);
}

// ---------------------------------------------------------------------------
// GRU gate GEMMs: gi = xbuf(32x1024) @ W_ih^T (+b_ih), gh = hidden(32x512) @ W_hh^T (+b_hh)
// Wave tile = 32 rows x 32 cols.  48 + 48 tiles -> 12 blocks x 8 waves.
// hidden lives in sbuf[:, :512] (row stride DH).
// ---------------------------------------------------------------------------
__global__ void k_gates(const float* __restrict__ xbuf, const float* __restrict__ sbuf,
                        const float* __restrict__ W_ih, const float* __restrict__ b_ih,
                        const float* __restrict__ W_hh, const float* __restrict__ b_hh,
                        float* __restrict__ gi, float* __restrict__ gh) {
    int wave = threadIdx.x >> 5, lane = threadIdx.x & 31;
    int tile = blockIdx.x * 8 + wave;          // 0..95 (uniform per wave)
    const float *A, *W, *bias; float* out; int K, sW, nb;
    if (tile < 48) { nb = tile * 32;        A = xbuf; W = W_ih + (long)nb * DH;  out = gi; bias = b_ih; K = DH;  sW = DH;  }
    else           { nb = (tile - 48) * 32; A = sbuf; W = W_hh + (long)nb * HID; out = gh; bias = b_hh; K = HID; sW = HID; }
    const float* W1 = W + 16 * sW;
    int r = lane & 15, kh = (lane >> 4) * 2;
    v8f c00 = {}, c01 = {}, c10 = {}, c11 = {};
#pragma unroll 4
    for (int k = 0; k < K; k += 4) {
        v2f a0 = *(const v2f*)(A  + r * DH + k + kh);
        v2f a1 = *(const v2f*)(A  + (r + 16) * DH + k + kh);
        v2f b0 = *(const v2f*)(W  + r * sW + k + kh);
        v2f b1 = *(const v2f*)(W1 + r * sW + k + kh);
        c00 = wmma4(a0, b0, c00);
        c10 = wmma4(a1, b0, c10);
        c01 = wmma4(a0, b1, c01);
        c11 = wmma4(a1, b1, c11);
    }
    float bv0 = bias[nb + r];
    float bv1 = bias[nb + 16 + r];
#pragma unroll
    for (int i = 0; i < 8; ++i) { c00[i] += bv0; c10[i] += bv0; c01[i] += bv1; c11[i] += bv1; }
    store_tile16(out + nb,      G3, c00, lane, 0);
    store_tile16(out + nb,      G3, c10, lane, 16);
    store_tile16(out + nb + 16, G3, c01, lane, 0);
    store_tile16(out + nb + 16, G3, c11, lane, 16);
}

// ---------------------------------------------------------------------------
// GRU elementwise update: hidden (in sbuf[:, :512]) <- (1-z)*n + z*hidden
// ---------------------------------------------------------------------------
__global__ void k_gru_update(const float* __restrict__ gi, const float* __restrict__ gh,
                             float* __restrict__ sbuf) {
    int idx = blockIdx.x * 256 + threadIdx.x;   // 0..16383
    int b = idx >> 9, j = idx & 511;
    long gb = (long)b * G3;
    float ir = gi[gb + j], iz = gi[gb + HID + j], in = gi[gb + 2 * HID + j];
    float hr = gh[gb + j], hz = gh[gb + HID + j], hn = gh[gb + 2 * HID + j];
    float rg = 1.0f / (1.0f + __expf(-(ir + hr)));
    float zg = 1.0f / (1.0f + __expf(-(iz + hz)));
    float ng = tanhf(in + rg * hn);
    float* hp = sbuf + (long)b * DH + j;
    float h  = *hp;
    *hp = (1.0f - zg) * ng + zg * h;
}

// ---------------------------------------------------------------------------
// Output projection: out[b*32+step, :] = sbuf(32x1024) @ W_out^T + b_out
// N=32000 -> 1000 wave-tiles of 32x32; 125 blocks x 8 waves.
// Per K-step: 2 A-frag + 2 B-frag loads feed 4 WMMAs (1.0 load/WMMA).
// ---------------------------------------------------------------------------
__global__ void k_score(const float* __restrict__ sbuf, const float* __restrict__ W_out,
                        const float* __restrict__ b_out, float* __restrict__ out, int step) {
    int wave = threadIdx.x >> 5, lane = threadIdx.x & 31;
    int tile = blockIdx.x * 8 + wave;          // 0..999
    int nb = tile * 32;
    const float* W0 = W_out + (long)nb * DH;
    const float* W1 = W0 + 16 * DH;
    int r = lane & 15, kh = (lane >> 4) * 2;
    v8f c00 = {}, c01 = {}, c10 = {}, c11 = {};
#pragma unroll 4
    for (int k = 0; k < DH; k += 4) {
        v2f a0 = *(const v2f*)(sbuf + r * DH + k + kh);
        v2f a1 = *(const v2f*)(sbuf + (r + 16) * DH + k + kh);
        v2f b0 = *(const v2f*)(W0 + r * DH + k + kh);
        v2f b1 = *(const v2f*)(W1 + r * DH + k + kh);
        c00 = wmma4(a0, b0, c00);
        c10 = wmma4(a1, b0, c10);
        c01 = wmma4(a0, b1, c01);
        c11 = wmma4(a1, b1, c11);
    }
    float bv0 = b_out[nb + r];
    float bv1 = b_out[nb + 16 + r];
    int n = lane & 15, mg = (lane >> 4) * 8;
#pragma unroll
    for (int i = 0; i < 8; ++i) {
        int m0 = mg + i;          // rows 0..15
        int m1 = 16 + mg + i;     // rows 16..31
        long r0 = ((long)m0 * LMAX + step) * VOCAB + nb + n;
        long r1 = ((long)m1 * LMAX + step) * VOCAB + nb + n;
        out[r0]      = c00[i] + bv0;
        out[r1]      = c10[i] + bv0;
        out[r0 + 16] = c01[i] + bv1;
        out[r1 + 16] = c11[i] + bv1;
    }
}

// ---------------------------------------------------------------------------
// Per-row: max, argmax, log-sum-exp over 32000. One block per batch row.
// ---------------------------------------------------------------------------
__global__ void k_rowreduce(const float* __restrict__ out, int step,
                            float* __restrict__ stats, int* __restrict__ decoded) {
    int b = blockIdx.x, tid = threadIdx.x;
    const float* row = out + ((long)b * LMAX + step) * VOCAB;
    __shared__ float smax[256]; __shared__ int sarg[256]; __shared__ float ssum[256];
    float m = -3.4e38f; int arg = 0;
    for (int j = tid; j < VOCAB; j += 256) { float v = row[j]; if (v > m) { m = v; arg = j; } }
    smax[tid] = m; sarg[tid] = arg; __syncthreads();
    for (int s = 128; s > 0; s >>= 1) {
        if (tid < s) {
            if (smax[tid + s] > smax[tid] ||
                (smax[tid + s] == smax[tid] && sarg[tid + s] < sarg[tid])) {
                smax[tid] = smax[tid + s]; sarg[tid] = sarg[tid + s];
            }
        }
        __syncthreads();
    }
    float rowmax = smax[0];
    float sum = 0.0f;
    for (int j = tid; j < VOCAB; j += 256) sum += __expf(row[j] - rowmax);
    ssum[tid] = sum; __syncthreads();
    for (int s = 128; s > 0; s >>= 1) {
        if (tid < s) ssum[tid] += ssum[tid + s];
        __syncthreads();
    }
    if (tid == 0) {
        stats[b * 2]     = rowmax;
        stats[b * 2 + 1] = logf(ssum[0]);
        decoded[b]       = sarg[0];
    }
}

// logp = score - (max + log sum)
__global__ void k_normalize(float* __restrict__ out, int step, const float* __restrict__ stats) {
    long idx = (long)blockIdx.x * 256 + threadIdx.x;  // 0 .. 32*32000-1
    int b = (int)(idx / VOCAB);
    int j = (int)(idx % VOCAB);
    out[((long)b * LMAX + step) * VOCAB + j] -= stats[b * 2] + stats[b * 2 + 1];
}

// ---------------------------------------------------------------------------
// Embedding gather + attention. One block per batch row, 128 threads (= T).
// Writes xbuf[:, :512] = emb[decoded], xbuf/sbuf[:, 512:] = new ctx.
// ---------------------------------------------------------------------------
__global__ void k_embed_attn(const float* __restrict__ emb, const int* __restrict__ decoded,
                             const float* __restrict__ energies, const float* __restrict__ enc,
                             float* __restrict__ xbuf, float* __restrict__ sbuf) {
    int b = blockIdx.x, t = threadIdx.x;
    __shared__ float alpha[TLEN]; __shared__ float red[TLEN];
    // gather next-step embedding
    int tok = decoded[b];
    for (int j = t; j < DIM; j += TLEN)
        xbuf[(long)b * DH + j] = emb[(long)tok * DIM + j];
    // attention energy: ae[t] = energies[b,t,:] . hidden[b,:]
    const float* hid  = sbuf + (long)b * DH;
    const float* erow = energies + ((long)b * TLEN + t) * HID;
    float ae = 0.0f;
#pragma unroll 4
    for (int h = 0; h < HID; ++h) ae += erow[h] * hid[h];
    alpha[t] = ae; red[t] = ae; __syncthreads();
    for (int s = TLEN / 2; s > 0; s >>= 1) {
        if (t < s) red[t] = fmaxf(red[t], red[t + s]);
        __syncthreads();
    }
    float mx = red[0]; __syncthreads();
    float e = __expf(ae - mx);
    red[t] = e; __syncthreads();
    for (int s = TLEN / 2; s > 0; s >>= 1) {
        if (t < s) red[t] += red[t + s];
        __syncthreads();
    }
    float inv = 1.0f / red[0];
    alpha[t] = e * inv; __syncthreads();
    // ctx[h] = sum_t alpha[t] * enc[b,t,h]
    for (int h = t; h < HID; h += TLEN) {
        float acc = 0.0f;
        for (int tt = 0; tt < TLEN; ++tt)
            acc += alpha[tt] * enc[((long)b * TLEN + tt) * HID + h];
        xbuf[(long)b * DH + HID + h] = acc;
        sbuf[(long)b * DH + HID + h] = acc;
    }
}

// init: xbuf = [emb[inputs[b]], ctx0], sbuf = [0, ctx0]
__global__ void k_init(const int* __restrict__ inputs, const float* __restrict__ context,
                       const float* __restrict__ emb,
                       float* __restrict__ xbuf, float* __restrict__ sbuf) {
    int idx = blockIdx.x * 256 + threadIdx.x;   // 0 .. 32*1024-1
    int b = idx >> 10, j = idx & 1023;
    if (j < DIM) {
        xbuf[idx] = emb[(long)inputs[b] * DIM + j];
        sbuf[idx] = 0.0f;
    } else {
        float c = context[(long)b * HID + (j - DIM)];
        xbuf[idx] = c;
        sbuf[idx] = c;
    }
}

extern "C" void kernel_launch(void* const* d_in, const int* in_sizes, int n_in,
                              void* d_out, int out_size, void* d_ws, size_t ws_size,
                              hipStream_t stream) {
    const int*   inputs  = (const int*)  d_in[0];
    const float* context = (const float*)d_in[1];
    // d_in[2] = max_len (compile-time LMAX=32)
    const float* enc     = (const float*)d_in[3];
    const float* emb     = (const float*)d_in[4];
    const float* W_ih    = (const float*)d_in[5];
    const float* W_hh    = (const float*)d_in[6];
    const float* b_ih    = (const float*)d_in[7];
    const float* b_hh    = (const float*)d_in[8];
    const float* W_out   = (const float*)d_in[9];
    const float* b_out   = (const float*)d_in[10];
    const float* W_attn  = (const float*)d_in[11];
    const float* b_attn  = (const float*)d_in[12];
    float* out = (float*)d_out;

    // workspace carve-up (floats)
    float* ws       = (float*)d_ws;
    float* energies = ws;                                   // 32*128*512
    float* xbuf     = energies + (long)BATCH * TLEN * HID;  // 32*1024
    float* sbuf     = xbuf + BATCH * DH;                    // 32*1024
    float* gi       = sbuf + BATCH * DH;                    // 32*1536
    float* gh       = gi + BATCH * G3;                      // 32*1536
    float* stats    = gh + BATCH * G3;                      // 32*2
    int*   decoded  = (int*)(stats + BATCH * 2);            // 32

    k_init    <<<BATCH * DH / 256, 256, 0, stream>>>(inputs, context, emb, xbuf, sbuf);
    k_energies<<<(BATCH * TLEN / 32) * (HID / 32) / 8, 256, 0, stream>>>(enc, W_attn, b_attn, energies);

    for (int step = 0; step < LMAX; ++step) {
        k_gates     <<<(2 * G3 / 32) / 8, 256, 0, stream>>>(xbuf, sbuf, W_ih, b_ih, W_hh, b_hh, gi, gh);
        k_gru_update<<<BATCH * HID / 256, 256, 0, stream>>>(gi, gh, sbuf);
        k_score     <<<(VOCAB / 32) / 8, 256, 0, stream>>>(sbuf, W_out, b_out, out, step);
        k_rowreduce <<<BATCH, 256, 0, stream>>>(out, step, stats, decoded);
        k_normalize <<<BATCH * VOCAB / 256, 256, 0, stream>>>(out, step, stats);
        k_embed_attn<<<BATCH, TLEN, 0, stream>>>(emb, decoded, energies, enc, xbuf, sbuf);
    }
}